// CRF_31636729102671
// MI455X (gfx1250) — compile-verified
//
#include <hip/hip_runtime.h>
#include <stdint.h>

// Problem constants (from reference): B=128, S=512, T = NUM_LABELS+2 = 32.
#define B_N 128
#define S_N 512
#define T_N 32
#define START_I 30
#define END_I 31
#define RING 16   // LDS ring slots for streamed feature rows
#define DEPTH 12  // async loads kept in flight (RING-4: no slot collision)

// gfx1250 async global->LDS copy of one dword per lane (ASYNCcnt-tracked).
// Inline asm so it is portable across ROCm7.2/clang-22 and clang-23 toolchains.
__device__ __forceinline__ void async_row32(const float* g, float* lds, int lane) {
  unsigned            lds_off = (unsigned)(uintptr_t)(lds + lane);       // low 32b = LDS offset
  unsigned long long  gaddr   = (unsigned long long)(uintptr_t)(g + lane);
  asm volatile("global_load_async_to_lds_b32 %0, %1, off"
               :: "v"(lds_off), "v"(gaddr)
               : "memory");
}
__device__ __forceinline__ void wait_async_le11(void) {
  asm volatile("s_wait_asynccnt 11" ::: "memory");
}

__global__ __launch_bounds__(32) void viterbi_crf_kernel(
    const float* __restrict__ feats,   // (B,S,T)
    const float* __restrict__ trans,   // (T,T)
    const int*   __restrict__ mask,    // (B,S)
    float*       __restrict__ out)     // (B,S) decoded labels as float
{
  __shared__ unsigned char bpL[S_N * T_N];    // 16 KB backpointers (back_points layout)
  __shared__ float         stage[RING * T_N]; // 2 KB streamed feature rows
  __shared__ unsigned char dec[S_N];          // 512 B decoded path

  const int lane = threadIdx.x;               // state j
  const int b    = blockIdx.x;                // sequence
  const float* fb = feats + (size_t)b * S_N * T_N;

  // Transition rows in registers: r[i] (lane j holds trans[i][j]). 32 VGPRs.
  float r[T_N];
  #pragma unroll
  for (int i = 0; i < T_N; ++i) r[i] = trans[i * T_N + lane];
  const float tEnd = trans[lane * T_N + END_I];   // lane i holds trans[i][END]

  // length[b] = sum(mask[b,:]) via per-lane partials + xor-butterfly add.
  int acc = 0;
  #pragma unroll
  for (int k = 0; k < S_N / 32; ++k) acc += mask[(size_t)b * S_N + k * 32 + lane];
  #pragma unroll
  for (int off = 16; off > 0; off >>= 1) acc += __shfl_xor(acc, off, 32);
  const int len = acc;                            // uniform, >= 1

  // t = 0: init_partition[j] = feats[b,0,j] + trans[START][j]
  float p = fb[lane] + r[START_I];
  float last_part = p;                            // valid if len == 1

  // Prologue: fill the async pipeline with rows t = 1..DEPTH.
  #pragma unroll
  for (int k = 1; k <= DEPTH; ++k)
    async_row32(fb + (size_t)k * T_N, stage + (k & (RING - 1)) * T_N, lane);

  // -------- forward scan --------
  for (int t = 1; t < S_N; ++t) {
    wait_async_le11();                             // oldest of 12 in flight is done
    const float f = stage[(t & (RING - 1)) * T_N + lane];

    // Leaves: v[i] = p[i] + trans[i][j] (32 independent readlane+add).
    float val[T_N];
    int   ind[T_N];
    #pragma unroll
    for (int i = 0; i < T_N; ++i) {
      float s = __uint_as_float(__builtin_amdgcn_readlane(__float_as_uint(p), i));
      val[i] = s + r[i];
      ind[i] = i;
    }
    // Adjacent-pair tournament: left block always has smaller indices, and
    // left wins ties => exact jnp.argmax (first-max) semantics.
    // Critical path: 5 levels instead of a 32-long serial cmp chain.
    #pragma unroll
    for (int n = T_N; n > 1; n >>= 1) {
      #pragma unroll
      for (int k = 0; k < (n >> 1); ++k) {
        bool rgt = val[2 * k + 1] > val[2 * k];
        val[k] = rgt ? val[2 * k + 1] : val[2 * k];
        ind[k] = rgt ? ind[2 * k + 1] : ind[2 * k];
      }
    }
    const float cur = val[0];
    const int   bp  = ind[0];

    p = cur + f;
    if (t == len - 1) last_part = p;               // partition_history[last_pos]

    // back_points[t-1] = masked bp (0 where t >= len)
    bpL[(t - 1) * T_N + lane] = (t < len) ? (unsigned char)bp : (unsigned char)0;

    // keep exactly DEPTH async loads in flight (clamp source past end)
    const int tt  = t + DEPTH;
    const int src = (tt < S_N) ? tt : (S_N - 1);
    async_row32(fb + (size_t)src * T_N, stage + (tt & (RING - 1)) * T_N, lane);
  }
  bpL[(S_N - 1) * T_N + lane] = 0;                 // back_points[S-1] = 0

  // pointer = argmax_i(last_part[i] + trans[i][END]); min-index tie-break.
  float v  = last_part + tEnd;
  int  idx = lane;
  #pragma unroll
  for (int off = 16; off > 0; off >>= 1) {
    float ov = __shfl_xor(v, off, 32);
    int   oi = __shfl_xor(idx, off, 32);
    if (ov > v || (ov == v && oi < idx)) { v = ov; idx = oi; }
  }
  const int pointer = idx;                         // uniform

  // back_points[last_pos][:, :] = pointer (broadcast over T)
  bpL[(len - 1) * T_N + lane] = (unsigned char)pointer;
  __syncthreads();

  // -------- backtrace (serial LDS pointer chase, lane 0) --------
  if (lane == 0) {
    int ptr = pointer;
    dec[S_N - 1] = (unsigned char)ptr;
    for (int s = S_N - 2; s >= 0; --s) {
      ptr = bpL[s * T_N + ptr];
      dec[s] = (unsigned char)ptr;
    }
  }
  __syncthreads();

  // Coalesced float output.
  #pragma unroll
  for (int k = 0; k < S_N / 32; ++k) {
    const int i2 = k * 32 + lane;
    out[(size_t)b * S_N + i2] = (float)dec[i2];
  }
}

extern "C" void kernel_launch(void* const* d_in, const int* in_sizes, int n_in,
                              void* d_out, int out_size, void* d_ws, size_t ws_size,
                              hipStream_t stream) {
  (void)in_sizes; (void)n_in; (void)d_ws; (void)ws_size; (void)out_size;
  const float* feats = (const float*)d_in[0];
  const float* trans = (const float*)d_in[1];
  const int*   mask  = (const int*)d_in[2];
  float*       out   = (float*)d_out;

  hipLaunchKernelGGL(viterbi_crf_kernel, dim3(B_N), dim3(32), 0, stream,
                     feats, trans, mask, out);
}